// DiffeomorphicTransform_28226525070088
// MI455X (gfx1250) — compile-verified
//
#include <hip/hip_runtime.h>

// DiffeomorphicTransform (scaling-and-squaring), MI455X / gfx1250.
// flow = velocity / 2^7; repeat 7x: flow += trilinear(flow, x + 100*flow)
// Pure data-dependent trilinear gather: no matmul structure (per-voxel unique
// weights AND operands) -> WMMA/TDM inapplicable. Field (39.3 MB) is fully
// L2-resident (192 MB), so this is an L2-gather-bandwidth kernel:
// branch-free masks, clamped indices, 27 unconditional global_load_b32 per
// voxel in deep clauses, wave32 blocks, global_prefetch_b8 on the linear
// stream. 3D launch removes all integer div/mod from the address path.

namespace {
constexpr int      kD = 128, kH = 160, kW = 160;
constexpr unsigned kN = (unsigned)kD * kH * kW;   // 3,276,800 voxels per channel
constexpr float    kRangeFlow = 100.0f;
constexpr unsigned kPFDist = 16384;               // ~64KB ahead on self-stream
}

// block = (32, 8, 1): one wave spans 32 consecutive x (coalesced), 8 waves/WG.
// grid  = (W/32, H/8, D) = (5, 20, 128): exact cover, no bounds checks.
__global__ __launch_bounds__(256) void diffeo_step(
    const float* __restrict__ fin, float* __restrict__ fout, float scale)
{
    const unsigned x = blockIdx.x * 32u + threadIdx.x;   // 0..159
    const unsigned y = blockIdx.y * 8u  + threadIdx.y;   // 0..159
    const unsigned z = blockIdx.z;                       // 0..127
    const unsigned idx = (z * (unsigned)kH + y) * (unsigned)kW + x;

    // gfx1250 global_prefetch_b8: warm L2 ahead on the coalesced self-stream.
    // Branch-free in-bounds clamp (select, no EXEC diamond).
    {
        const unsigned pfc = idx + kPFDist;
        const unsigned pf  = (pfc < kN) ? pfc : idx;
        __builtin_prefetch(fin + pf, 0, 1);
    }

    // Stored field values (actual flow = scale * stored; scale==1 except step 1
    // where fin==velocity and scale==2^-7, an exact power of two).
    const float fx = fin[idx];
    const float fy = fin[kN + idx];
    const float fz = fin[2u * kN + idx];

    // Sample coordinate: grid-normalize + denormalize cancels to x + 100*flow.
    const float ix = fmaf(kRangeFlow * scale, fx, (float)x);
    const float iy = fmaf(kRangeFlow * scale, fy, (float)y);
    const float iz = fmaf(kRangeFlow * scale, fz, (float)z);

    const float x0f = floorf(ix), y0f = floorf(iy), z0f = floorf(iz);
    const float x1f = x0f + 1.0f, y1f = y0f + 1.0f, z1f = z0f + 1.0f;

    float wx1 = ix - x0f, wy1 = iy - y0f, wz1 = iz - z0f;
    float wx0 = 1.0f - wx1, wy0 = 1.0f - wy1, wz0 = 1.0f - wz1;

    // Zero-padding: fold per-axis OOB masks into the 1-D weights (branch-free).
    wx0 = (x0f >= 0.0f && x0f <= (float)(kW - 1)) ? wx0 : 0.0f;
    wx1 = (x1f >= 0.0f && x1f <= (float)(kW - 1)) ? wx1 : 0.0f;
    wy0 = (y0f >= 0.0f && y0f <= (float)(kH - 1)) ? wy0 : 0.0f;
    wy1 = (y1f >= 0.0f && y1f <= (float)(kH - 1)) ? wy1 : 0.0f;
    wz0 = (z0f >= 0.0f && z0f <= (float)(kD - 1)) ? wz0 : 0.0f;
    wz1 = (z1f >= 0.0f && z1f <= (float)(kD - 1)) ? wz1 : 0.0f;

    // Clamp in float first (safe for arbitrarily large displacements), then cvt.
    const unsigned xi0 = (unsigned)(int)fminf(fmaxf(x0f, 0.0f), (float)(kW - 1));
    const unsigned xi1 = (unsigned)(int)fminf(fmaxf(x1f, 0.0f), (float)(kW - 1));
    const unsigned yi0 = (unsigned)(int)fminf(fmaxf(y0f, 0.0f), (float)(kH - 1));
    const unsigned yi1 = (unsigned)(int)fminf(fmaxf(y1f, 0.0f), (float)(kH - 1));
    const unsigned zi0 = (unsigned)(int)fminf(fmaxf(z0f, 0.0f), (float)(kD - 1));
    const unsigned zi1 = (unsigned)(int)fminf(fmaxf(z1f, 0.0f), (float)(kD - 1));

    // Row base offsets for the 4 (z,y) pairs.
    const unsigned r00 = (zi0 * (unsigned)kH + yi0) * (unsigned)kW;
    const unsigned r01 = (zi0 * (unsigned)kH + yi1) * (unsigned)kW;
    const unsigned r10 = (zi1 * (unsigned)kH + yi0) * (unsigned)kW;
    const unsigned r11 = (zi1 * (unsigned)kH + yi1) * (unsigned)kW;

    const float wzy00 = wz0 * wy0, wzy01 = wz0 * wy1;
    const float wzy10 = wz1 * wy0, wzy11 = wz1 * wy1;

    float a0 = 0.0f, a1 = 0.0f, a2 = 0.0f;
    auto corner = [&](unsigned ci, float w) {
        a0 = fmaf(fin[ci], w, a0);
        a1 = fmaf(fin[kN + ci], w, a1);
        a2 = fmaf(fin[2u * kN + ci], w, a2);
    };
    corner(r00 + xi0, wzy00 * wx0);
    corner(r00 + xi1, wzy00 * wx1);
    corner(r01 + xi0, wzy01 * wx0);
    corner(r01 + xi1, wzy01 * wx1);
    corner(r10 + xi0, wzy10 * wx0);
    corner(r10 + xi1, wzy10 * wx1);
    corner(r11 + xi0, wzy11 * wx0);
    corner(r11 + xi1, wzy11 * wx1);

    // new_flow = scale*(stored + gathered); exact vs reference since scale is
    // 1.0 (steps 2..7) or 2^-7 (step 1, folded velocity scaling).
    fout[idx]           = scale * (fx + a0);
    fout[kN + idx]      = scale * (fy + a1);
    fout[2u * kN + idx] = scale * (fz + a2);
}

extern "C" void kernel_launch(void* const* d_in, const int* in_sizes, int n_in,
                              void* d_out, int out_size, void* d_ws, size_t ws_size,
                              hipStream_t stream) {
    (void)in_sizes; (void)n_in; (void)out_size; (void)ws_size;
    const float* vel = (const float*)d_in[0];   // [1,3,128,160,160] fp32
    // d_in[1] (sample_grid) is the identity mesh; its normalization round-trips
    // to ix = x + 100*flow_x, so it is computed analytically.
    float* out = (float*)d_out;                 // ping buffer (final result here)
    float* ws  = (float*)d_ws;                  // pong buffer (needs 3*kN floats)

    const dim3 grid(kW / 32, kH / 8, kD);       // (5, 20, 128), exact cover
    const dim3 block(32, 8, 1);                 // 8 wave32s per workgroup
    const float s0 = 1.0f / 128.0f;             // 2^-TIME_STEP, exact

    // 7 steps (odd): vel -> out -> ws -> out -> ws -> out -> ws -> out.
    diffeo_step<<<grid, block, 0, stream>>>(vel, out, s0);   // step 1 (scale folded)
    diffeo_step<<<grid, block, 0, stream>>>(out, ws, 1.0f);  // step 2
    diffeo_step<<<grid, block, 0, stream>>>(ws, out, 1.0f);  // step 3
    diffeo_step<<<grid, block, 0, stream>>>(out, ws, 1.0f);  // step 4
    diffeo_step<<<grid, block, 0, stream>>>(ws, out, 1.0f);  // step 5
    diffeo_step<<<grid, block, 0, stream>>>(out, ws, 1.0f);  // step 6
    diffeo_step<<<grid, block, 0, stream>>>(ws, out, 1.0f);  // step 7 -> d_out
}